// DiffusionVIProcess_4054449128014
// MI455X (gfx1250) — compile-verified
//
#include <hip/hip_runtime.h>
#include <math.h>

typedef __attribute__((ext_vector_type(2))) float v2f;
typedef __attribute__((ext_vector_type(8))) float v8f;

namespace {
constexpr int T_STEPS = 64;
constexpr int S_DIM   = 16;
constexpr int B_DIM   = 2048;
constexpr int ZD      = 64;
constexpr int ROWS    = S_DIM * B_DIM;        // 32768 flattened (S,B) rows
constexpr int ROWS_PER_WG = 64;               // 4 waves x 16 rows
constexpr int NBLOCKS = ROWS / ROWS_PER_WG;   // 512
constexpr int ZS_STRIDE = 68;                 // padded LDS row stride (floats):
                                              // even (8B-aligned v2f) + conflict-free
}

// One workgroup = 4 waves; each wave owns a private 16x64 tile of z for the
// entire T-step scan. z lives in registers (C/D layout) + a private LDS slice
// (row-major, padded) used only to re-read it in WMMA A-operand layout.
__global__ __launch_bounds__(128)
void vi_scan_kernel(const float* __restrict__ z0,
                    const float* __restrict__ eps,
                    const float* __restrict__ ctxm,
                    const float* __restrict__ W,
                    const float* __restrict__ Wc,
                    const float* __restrict__ bmat,
                    const float* __restrict__ sigmas,
                    float* __restrict__ out,        // [0]=log_w slot, +1 = z_samples
                    float* __restrict__ partials)   // [NBLOCKS]
{
  __shared__ float sW[ZD * ZD];            // 16 KB
  __shared__ float sb[T_STEPS * ZD];       // 16 KB
  __shared__ float ssig[T_STEPS];
  __shared__ float zs[4 * 16 * ZS_STRIDE]; // 4 private wave slices, ~17 KB
  __shared__ float red[128];

  const int tid  = threadIdx.x;
  const int wave = tid >> 5;
  const int lane = tid & 31;
  const int half = lane >> 4;   // 0: lanes 0-15, 1: lanes 16-31
  const int ln16 = lane & 15;

  for (int i = tid; i < ZD * ZD; i += 128)      sW[i] = W[i];
  for (int i = tid; i < T_STEPS * ZD; i += 128) sb[i] = bmat[i];
  if (tid < T_STEPS) ssig[tid] = sigmas[tid];
  __syncthreads();

  const int row0 = blockIdx.x * ROWS_PER_WG + wave * 16;
  float* zsw = &zs[wave * 16 * ZS_STRIDE];

  const float dt = 1.0f / (float)T_STEPS;
  const float sqrt_dt = sqrtf(dt);

  v2f  a_sl[16];                 // A-operand slices, 16x4 f32 each
  v8f  gprev[4];                 // g = z @ W, carried across steps (D layout)
  v8f  ctxd[4];                  // ctx_drift tile (D layout)
  float zcur[4][8];              // current z (D layout)
  float esv[4][8];               // std*eps for this step
  float acc_lw = 0.0f;

  // ---------- ctx_drift tile: context[brow0:brow0+16,:] @ Wc (WMMA) ----------
  const int brow0 = row0 % B_DIM;   // same batch index per row modulo S
  #pragma unroll
  for (int ks = 0; ks < 16; ++ks) {
    const int k = ks * 4 + 2 * half;                  // A: M=ln16, K=k,k+1
    a_sl[ks] = *(const v2f*)&ctxm[(size_t)(brow0 + ln16) * ZD + k];
  }
  #pragma unroll
  for (int nt = 0; nt < 4; ++nt) {
    v8f acc = {0.f, 0.f, 0.f, 0.f, 0.f, 0.f, 0.f, 0.f};
    const int n = nt * 16 + ln16;
    #pragma unroll
    for (int ks = 0; ks < 16; ++ks) {
      const int k = ks * 4 + 2 * half;                // B: N=ln16 half-col, K=k,k+1
      v2f bb;
      bb.x = Wc[(size_t)k * ZD + n];
      bb.y = Wc[(size_t)(k + 1) * ZD + n];
      acc = __builtin_amdgcn_wmma_f32_16x16x4_f32(false, a_sl[ks], false, bb,
                                                  (short)0, acc, false, false);
    }
    ctxd[nt] = acc;
  }

  // ---------- init with z0: registers + z_samples[0] + LDS slice ----------
  #pragma unroll
  for (int nt = 0; nt < 4; ++nt) {
    #pragma unroll
    for (int r = 0; r < 8; ++r) {
      const int lrow = r + 8 * half;
      const int row  = row0 + lrow;
      const int col  = nt * 16 + ln16;
      const float v  = z0[(size_t)row * ZD + col];
      zcur[nt][r] = v;
      acc_lw += 0.5f * v * v;                               // +0.5*z0^2 boundary
      out[1 + (size_t)row * ZD + col] = v;                  // z_samples[0]
      zsw[lrow * ZS_STRIDE + col] = v;
    }
  }

  // g_prev = z0 @ W
  #pragma unroll
  for (int ks = 0; ks < 16; ++ks)
    a_sl[ks] = *(const v2f*)&zsw[ln16 * ZS_STRIDE + ks * 4 + 2 * half];
  #pragma unroll
  for (int nt = 0; nt < 4; ++nt) {
    v8f acc = {0.f, 0.f, 0.f, 0.f, 0.f, 0.f, 0.f, 0.f};
    const int n = nt * 16 + ln16;
    #pragma unroll
    for (int ks = 0; ks < 16; ++ks) {
      const int k = ks * 4 + 2 * half;
      v2f bb; bb.x = sW[k * ZD + n]; bb.y = sW[(k + 1) * ZD + n];
      acc = __builtin_amdgcn_wmma_f32_16x16x4_f32(false, a_sl[ks], false, bb,
                                                  (short)0, acc, false, false);
    }
    gprev[nt] = acc;
  }

  // ---------- sequential T-step scan (no barriers: LDS slice is private) ----
  for (int t = 0; t < T_STEPS; ++t) {
    const float sig   = ssig[t];
    const float stdv  = sig * sqrt_dt;
    const float istd2 = 1.0f / (stdv * stdv);
    float bt[4];
    #pragma unroll
    for (int nt = 0; nt < 4; ++nt) bt[nt] = sb[t * ZD + nt * 16 + ln16];

    // z_new = z + (g_prev + ctx + b_t)*dt + std*eps ; stream eps in, z out
    #pragma unroll
    for (int nt = 0; nt < 4; ++nt) {
      #pragma unroll
      for (int r = 0; r < 8; ++r) {
        const int lrow = r + 8 * half;
        const int row  = row0 + lrow;
        const int col  = nt * 16 + ln16;
        const float e  = __builtin_nontemporal_load(
            &eps[((size_t)t * ROWS + row) * ZD + col]);
        const float es = stdv * e;
        const float zn = zcur[nt][r] +
            (gprev[nt][r] + ctxd[nt][r] + bt[nt]) * dt + es;
        zcur[nt][r] = zn;
        esv[nt][r]  = es;
        __builtin_nontemporal_store(zn,
            &out[1 + ((size_t)(t + 1) * ROWS + row) * ZD + col]);
        zsw[lrow * ZS_STRIDE + col] = zn;
      }
    }

    // re-read new z as WMMA A-operand slices
    #pragma unroll
    for (int ks = 0; ks < 16; ++ks)
      a_sl[ks] = *(const v2f*)&zsw[ln16 * ZS_STRIDE + ks * 4 + 2 * half];

    // g_new = z_new @ W ; accumulate log-weight:
    // diff = (g_new - g_prev - ctx)*dt - std*e   (b_t cancels exactly)
    // lw  += 0.5*((std*e)^2 - diff^2)/std^2
    #pragma unroll
    for (int nt = 0; nt < 4; ++nt) {
      v8f acc = {0.f, 0.f, 0.f, 0.f, 0.f, 0.f, 0.f, 0.f};
      const int n = nt * 16 + ln16;
      #pragma unroll
      for (int ks = 0; ks < 16; ++ks) {
        const int k = ks * 4 + 2 * half;
        v2f bb; bb.x = sW[k * ZD + n]; bb.y = sW[(k + 1) * ZD + n];
        acc = __builtin_amdgcn_wmma_f32_16x16x4_f32(false, a_sl[ks], false, bb,
                                                    (short)0, acc, false, false);
      }
      #pragma unroll
      for (int r = 0; r < 8; ++r) {
        const float diff =
            (acc[r] - gprev[nt][r] - ctxd[nt][r]) * dt - esv[nt][r];
        acc_lw += 0.5f * (esv[nt][r] * esv[nt][r] - diff * diff) * istd2;
      }
      gprev[nt] = acc;
    }
  }

  // boundary: -0.5*zT^2
  #pragma unroll
  for (int nt = 0; nt < 4; ++nt)
    #pragma unroll
    for (int r = 0; r < 8; ++r)
      acc_lw -= 0.5f * zcur[nt][r] * zcur[nt][r];

  // deterministic in-block reduction -> one partial per workgroup
  red[tid] = acc_lw;
  __syncthreads();
  if (tid == 0) {
    float s = 0.f;
    for (int i = 0; i < 128; ++i) s += red[i];
    partials[blockIdx.x] = s;
  }
}

// deterministic final reduction: log_w = sum(partials)/S
__global__ void vi_reduce_kernel(const float* __restrict__ partials,
                                 float* __restrict__ out) {
  if (threadIdx.x == 0 && blockIdx.x == 0) {
    float s = 0.f;
    for (int i = 0; i < NBLOCKS; ++i) s += partials[i];
    out[0] = s * (1.0f / (float)S_DIM);
  }
}

extern "C" void kernel_launch(void* const* d_in, const int* in_sizes, int n_in,
                              void* d_out, int out_size, void* d_ws, size_t ws_size,
                              hipStream_t stream) {
  const float* z0   = (const float*)d_in[0];
  const float* eps  = (const float*)d_in[1];
  const float* ctx  = (const float*)d_in[2];
  const float* W    = (const float*)d_in[3];
  const float* Wc   = (const float*)d_in[4];
  const float* b    = (const float*)d_in[5];
  const float* sig  = (const float*)d_in[6];
  float* out      = (float*)d_out;
  float* partials = (float*)d_ws;   // 512 floats of scratch

  vi_scan_kernel<<<NBLOCKS, 128, 0, stream>>>(z0, eps, ctx, W, Wc, b, sig,
                                              out, partials);
  vi_reduce_kernel<<<1, 32, 0, stream>>>(partials, out);
}